// GCN_42941083025466
// MI455X (gfx1250) — compile-verified
//
#include <hip/hip_runtime.h>

typedef __attribute__((ext_vector_type(2))) float v2f;
typedef __attribute__((ext_vector_type(8))) float v8f;

#define GCN_N     100000
#define GCN_DIN   128
#define GCN_DHID  128
#define GCN_DOUT  64

// ---------------------------------------------------------------- zero fill
__global__ void gcn_zero_f4(float4* __restrict__ p, int n4) {
    int i = blockIdx.x * blockDim.x + threadIdx.x;
    if (i < n4) p[i] = make_float4(0.f, 0.f, 0.f, 0.f);
}

// ---------------------------------------------------------------- fp32 WMMA GEMM
// C[M x NC] = (RELU ? relu(A) : A)[M x K] @ B[K x NC]
// One wave per 16x16 output tile; K-loop in steps of 4 via v_wmma_f32_16x16x4_f32.
template <int K, int NC, bool RELU>
__global__ void gcn_gemm_wmma(const float* __restrict__ A,
                              const float* __restrict__ B,
                              float* __restrict__ C, int M) {
    const int wave = threadIdx.x >> 5;
    const int lane = threadIdx.x & 31;
    const int tile = blockIdx.x * (blockDim.x >> 5) + wave;
    const int ntiles = NC / 16;
    const int m0 = (tile / ntiles) * 16;
    const int n0 = (tile % ntiles) * 16;
    if (m0 >= M) return;               // wave-uniform; EXEC stays all-ones inside

    const int row = lane & 15;         // M for A-load, N for B-load / C-store
    const int kb  = (lane >> 4) << 1;  // K-pair select: 0 or 2
    const float* __restrict__ arow = A + (size_t)(m0 + row) * K;

    v8f acc = {};
#pragma unroll 4
    for (int k = 0; k < K; k += 4) {
        const int kk = k + kb;
        float a0 = arow[kk];
        float a1 = arow[kk + 1];
        if (RELU) { a0 = fmaxf(a0, 0.f); a1 = fmaxf(a1, 0.f); }
        v2f a; a.x = a0; a.y = a1;
        v2f b;
        b.x = B[(size_t)kk       * NC + n0 + row];
        b.y = B[(size_t)(kk + 1) * NC + n0 + row];
        // 8 args: (neg_a, A, neg_b, B, c_mod, C, reuse_a, reuse_b)
        acc = __builtin_amdgcn_wmma_f32_16x16x4_f32(false, a, false, b,
                                                    (short)0, acc, false, false);
    }

    const int moff = (lane >> 4) << 3; // lanes 16-31 hold M = r + 8
#pragma unroll
    for (int r = 0; r < 8; ++r)
        C[(size_t)(m0 + r + moff) * NC + n0 + row] = acc[r];
}

// ---------------------------------------------------------------- edge scatter
// out[src[e], :] += val[e] * xw[dst[e], :]   (C channels, float4 per thread)
template <int C>
__global__ void gcn_edge_agg(const float* __restrict__ xw,
                             const int* __restrict__ src,
                             const int* __restrict__ dst,
                             const float* __restrict__ val,
                             float* __restrict__ out, int E) {
    constexpr int G = C / 4;
    long tid = (long)blockIdx.x * blockDim.x + threadIdx.x;
    if (tid >= (long)E * G) return;
    const int e = (int)(tid >> (C == 128 ? 5 : 4));
    const int g = (int)(tid & (G - 1));
    const int s = src[e];
    const int d = dst[e];
    const float v = val[e];
    const float4 m = ((const float4*)(xw + (size_t)d * C))[g];
    float* o = out + (size_t)s * C + g * 4;
    atomicAdd(o + 0, v * m.x);
    atomicAdd(o + 1, v * m.y);
    atomicAdd(o + 2, v * m.z);
    atomicAdd(o + 3, v * m.w);
}

// ---------------------------------------------------------------- relu + softmax (64 ch)
__global__ void gcn_relu_softmax64(float* __restrict__ out, int n) {
    const int wave = threadIdx.x >> 5;
    const int lane = threadIdx.x & 31;
    const int rowi = blockIdx.x * (blockDim.x >> 5) + wave;
    if (rowi >= n) return;
    float* p = out + (size_t)rowi * 64;
    float h0 = fmaxf(p[lane], 0.f);
    float h1 = fmaxf(p[lane + 32], 0.f);
    float m = fmaxf(h0, h1);
#pragma unroll
    for (int off = 16; off > 0; off >>= 1)
        m = fmaxf(m, __shfl_xor(m, off, 32));
    float e0 = __expf(h0 - m);
    float e1 = __expf(h1 - m);
    float s = e0 + e1;
#pragma unroll
    for (int off = 16; off > 0; off >>= 1)
        s += __shfl_xor(s, off, 32);
    const float inv = 1.0f / s;
    p[lane]      = e0 * inv;
    p[lane + 32] = e1 * inv;
}

// ---------------------------------------------------------------- launch
extern "C" void kernel_launch(void* const* d_in, const int* in_sizes, int n_in,
                              void* d_out, int out_size, void* d_ws, size_t ws_size,
                              hipStream_t stream) {
    const float* x    = (const float*)d_in[0];
    const float* W1   = (const float*)d_in[1];
    const float* W2   = (const float*)d_in[2];
    const int*   esrc = (const int*)d_in[3];
    const int*   edst = (const int*)d_in[4];
    const float* eval = (const float*)d_in[5];
    float* out = (float*)d_out;

    const int N = in_sizes[0] / GCN_DIN;   // 100000
    const int E = in_sizes[3];             // 1.7M

    float* buf0 = (float*)d_ws;                    // XW1, later reused for HW2
    float* buf1 = buf0 + (size_t)N * GCN_DHID;     // H accumulator

    // 1) zero layer-1 accumulator
    {
        int n4 = N * GCN_DHID / 4;
        gcn_zero_f4<<<(n4 + 255) / 256, 256, 0, stream>>>((float4*)buf1, n4);
    }
    // 2) XW1 = x @ W1
    {
        int tiles = (N / 16) * (GCN_DHID / 16);    // 50000
        gcn_gemm_wmma<GCN_DIN, GCN_DHID, false>
            <<<(tiles + 3) / 4, 128, 0, stream>>>(x, W1, buf0, N);
    }
    // 3) H = A_hat @ XW1
    {
        long total = (long)E * (GCN_DHID / 4);
        gcn_edge_agg<GCN_DHID>
            <<<(int)((total + 255) / 256), 256, 0, stream>>>(buf0, esrc, edst, eval, buf1, E);
    }
    // 4) zero output accumulator
    {
        int n4 = N * GCN_DOUT / 4;
        gcn_zero_f4<<<(n4 + 255) / 256, 256, 0, stream>>>((float4*)out, n4);
    }
    // 5) HW2 = relu(H) @ W2   (relu fused into A-load; reuse buf0)
    {
        int tiles = (N / 16) * (GCN_DOUT / 16);    // 25000
        gcn_gemm_wmma<GCN_DHID, GCN_DOUT, true>
            <<<(tiles + 3) / 4, 128, 0, stream>>>(buf1, W2, buf0, N);
    }
    // 6) out = A_hat @ HW2
    {
        long total = (long)E * (GCN_DOUT / 4);
        gcn_edge_agg<GCN_DOUT>
            <<<(int)((total + 255) / 256), 256, 0, stream>>>(buf0, esrc, edst, eval, out, E);
    }
    // 7) softmax(relu(out)) row-wise, in place
    gcn_relu_softmax64<<<(N + 7) / 8, 256, 0, stream>>>(out, N);
}